// DrBCEncoder_43946105373340
// MI455X (gfx1250) — compile-verified
//
#include <hip/hip_runtime.h>
#include <hip/hip_bf16.h>

// DrBC / GraphSAGE encoder for MI455X (gfx1250, wave32).
// Memory-bound: ~5 GB of edge gather/scatter traffic dominates (0.2 ms @ 23.3 TB/s).
// GEMMs (64x64, fp32) run on the WMMA unit with V_WMMA_F32_16X16X4_F32 to keep
// full fp32 precision matching the reference.

#define HID   64
#define HSTR  68          // LDS row stride (pad: row*68 % 64 banks = row*4 -> conflict-free)
#define LN_EPS 1e-5f

typedef __attribute__((ext_vector_type(2))) float v2f;
typedef __attribute__((ext_vector_type(8))) float v8f;

// ---------------------------------------------------------------- degree
__global__ __launch_bounds__(256) void deg_kernel(const int* __restrict__ dst,
                                                  float* __restrict__ deg, int nEdges) {
    int gid = blockIdx.x * blockDim.x + threadIdx.x;
    if (gid < nEdges) atomicAdd(&deg[dst[gid]], 1.0f);
}

// ------------------------------------------------- input projection + relu
// h[n][j] = relu(b[j] + sum_k x[n][k] * W[j][k]),  IN_DIM = 8
__global__ __launch_bounds__(256) void input_kernel(const float* __restrict__ x,
                                                    const float* __restrict__ W,
                                                    const float* __restrict__ b,
                                                    float* __restrict__ h, int nNodes) {
    int gid = blockIdx.x * blockDim.x + threadIdx.x;
    if (gid >= nNodes * HID) return;
    int node = gid >> 6, j = gid & 63;
    float acc = b[j];
#pragma unroll
    for (int k = 0; k < 8; ++k) acc += x[node * 8 + k] * W[j * 8 + k];
    h[gid] = acc > 0.0f ? acc : 0.0f;
}

// ------------------------------------------------------ edge gather/scatter
// thread = (edge, 4-float chunk); 16 consecutive lanes share one edge so the
// h[src] row read is a coalesced 256B burst; scatter via global_atomic_add_f32.
__global__ __launch_bounds__(256) void scatter_kernel(const float* __restrict__ h,
                                                      const int* __restrict__ src,
                                                      const int* __restrict__ dst,
                                                      float* __restrict__ nsum, int nEdges) {
    long gid = (long)blockIdx.x * blockDim.x + threadIdx.x;
    if (gid >= (long)nEdges * 16) return;
    int e = (int)(gid >> 4);
    int c = ((int)gid & 15) << 2;
    int s = src[e], d = dst[e];
    const float4 v = *(const float4*)(h + (long)s * HID + c);
    float* p = nsum + (long)d * HID + c;
    atomicAdd(p + 0, v.x);
    atomicAdd(p + 1, v.y);
    atomicAdd(p + 2, v.z);
    atomicAdd(p + 3, v.w);
}

// --------------------------------------------------- layer: WMMA GEMM + LN
// Block = 16 nodes, 128 threads = 4 waves. Wave w computes output cols
// [16w, 16w+16). S = h@Ws^T + neigh_mean@Wn^T + bias, then LayerNorm, ReLU,
// residual. fp32 WMMA: D = A(16x4) x B(4x16) + C, K=64 -> 16 steps x 2 mats.
__global__ __launch_bounds__(128) void layer_kernel(float* __restrict__ h,
                                                    const float* __restrict__ nsum,
                                                    const float* __restrict__ deg,
                                                    const float* __restrict__ Wself,
                                                    const float* __restrict__ Wneigh,
                                                    const float* __restrict__ bias,
                                                    const float* __restrict__ gamma,
                                                    const float* __restrict__ beta,
                                                    float* __restrict__ out, int nNodes) {
    __shared__ float lh[16 * HSTR];    // h tile (also residual source)
    __shared__ float lng[16 * HSTR];   // neigh_mean tile
    __shared__ float lws[64 * HSTR];   // W_self (row-major W[j][k])
    __shared__ float lwn[64 * HSTR];   // W_neigh
    __shared__ float ls[16 * HSTR];    // S tile for LayerNorm reduction
    __shared__ float mu[16], rstd[16];

    const int tid = threadIdx.x;
    const int node0 = blockIdx.x * 16;

    // Stage node tiles (1024 floats each -> 8/thread) with on-the-fly deg norm.
    for (int i = tid; i < 16 * HID; i += 128) {
        int m = i >> 6, k = i & 63;
        int node = node0 + m;
        if (node < nNodes) {
            float d = deg[node];
            float inv = d > 0.0f ? 1.0f / d : 0.0f;
            lh[m * HSTR + k]  = h[(long)node * HID + k];
            lng[m * HSTR + k] = nsum[(long)node * HID + k] * inv;
        } else {
            lh[m * HSTR + k] = 0.0f;
            lng[m * HSTR + k] = 0.0f;
        }
    }
    // Stage weights (4096 floats each -> 32/thread).
    for (int i = tid; i < 64 * HID; i += 128) {
        int j = i >> 6, k = i & 63;
        lws[j * HSTR + k] = Wself[i];
        lwn[j * HSTR + k] = Wneigh[i];
    }
    __syncthreads();

    const int lane  = tid & 31;
    const int nb    = (tid >> 5) * 16;      // wave's N-column base
    const int row   = lane & 15;            // A: M row / B: N col within tile
    const int khalf = (lane >> 4) << 1;     // lanes 16-31 hold K+2,K+3

    // out[m][nb+n] = sum_k h[m][k]*W[nb+n][k]  => B[k][n] = W[nb+n][k].
    // A lane<16: {A[row][k0],A[row][k0+1]}; lane>=16: {A[row][k0+2],A[row][k0+3]}.
    // B has the mirrored layout, and since B = W^T both fragments read
    // two consecutive K values at LDS row (nb+)row.
    v8f acc = {};
#pragma unroll
    for (int k0 = 0; k0 < HID; k0 += 4) {
        const int kb = k0 + khalf;
        v2f a_s = { lh[row * HSTR + kb],        lh[row * HSTR + kb + 1] };
        v2f b_s = { lws[(nb + row) * HSTR + kb], lws[(nb + row) * HSTR + kb + 1] };
        acc = __builtin_amdgcn_wmma_f32_16x16x4_f32(false, a_s, false, b_s,
                                                    (short)0, acc, false, false);
        v2f a_n = { lng[row * HSTR + kb],        lng[row * HSTR + kb + 1] };
        v2f b_n = { lwn[(nb + row) * HSTR + kb], lwn[(nb + row) * HSTR + kb + 1] };
        acc = __builtin_amdgcn_wmma_f32_16x16x4_f32(false, a_n, false, b_n,
                                                    (short)0, acc, false, false);
    }

    // C/D layout: VGPR v -> M=v (lanes 0-15) / M=v+8 (lanes 16-31), N = lane%16.
    {
        const int mh = (lane >> 4) * 8;
        const int ncol = nb + row;
        const float bc = bias[ncol];
#pragma unroll
        for (int v = 0; v < 8; ++v)
            ls[(v + mh) * HSTR + ncol] = acc[v] + bc;
    }
    __syncthreads();

    // LayerNorm stats: one thread per node row (reduce over all 64 dims).
    if (tid < 16) {
        float s = 0.0f, s2 = 0.0f;
        for (int c = 0; c < HID; ++c) {
            float v = ls[tid * HSTR + c];
            s += v; s2 += v * v;
        }
        float m = s * (1.0f / HID);
        float var = s2 * (1.0f / HID) - m * m;
        mu[tid] = m;
        rstd[tid] = rsqrtf(var + LN_EPS);
    }
    __syncthreads();

    // Apply gamma/beta + ReLU + residual; write out.
    for (int i = tid; i < 16 * HID; i += 128) {
        int m = i >> 6, c = i & 63;
        int node = node0 + m;
        if (node >= nNodes) continue;
        float val = (ls[m * HSTR + c] - mu[m]) * rstd[m] * gamma[c] + beta[c];
        float r = val > 0.0f ? val : 0.0f;
        out[(long)node * HID + c] = lh[m * HSTR + c] + r;
    }
}

// -------------------------------------------------------------- launcher
extern "C" void kernel_launch(void* const* d_in, const int* in_sizes, int n_in,
                              void* d_out, int out_size, void* d_ws, size_t ws_size,
                              hipStream_t stream) {
    const float* x        = (const float*)d_in[0];
    const int*   esrc     = (const int*)d_in[1];
    const int*   edst     = (const int*)d_in[2];
    const float* W_in     = (const float*)d_in[3];
    const float* b_in     = (const float*)d_in[4];
    const float* Ws_self  = (const float*)d_in[5];
    const float* Ws_neigh = (const float*)d_in[6];
    const float* biases   = (const float*)d_in[7];
    const float* gammas   = (const float*)d_in[8];
    const float* betas    = (const float*)d_in[9];

    const int nNodes = in_sizes[0] / 8;   // x is (n, 8)
    const int nEdges = in_sizes[1];

    // Workspace: h (n*64), neigh_sum (n*64), deg (n). ~51.6 MB for n=100k.
    float* h    = (float*)d_ws;
    float* nsum = h + (size_t)nNodes * HID;
    float* deg  = nsum + (size_t)nNodes * HID;

    hipMemsetAsync(deg, 0, (size_t)nNodes * sizeof(float), stream);
    deg_kernel<<<(nEdges + 255) / 256, 256, 0, stream>>>(edst, deg, nEdges);

    const long hElems = (long)nNodes * HID;
    input_kernel<<<(int)((hElems + 255) / 256), 256, 0, stream>>>(x, W_in, b_in, h, nNodes);

    const int nTiles = (nNodes + 15) / 16;       // 6250 for n=100k
    const long swork = (long)nEdges * 16;

    for (int l = 0; l < 3; ++l) {
        hipMemsetAsync(nsum, 0, (size_t)nNodes * HID * sizeof(float), stream);
        scatter_kernel<<<(int)((swork + 255) / 256), 256, 0, stream>>>(h, esrc, edst,
                                                                       nsum, nEdges);
        float* outp = (l == 2) ? (float*)d_out : h;   // last layer -> d_out directly
        layer_kernel<<<nTiles, 128, 0, stream>>>(h, nsum, deg,
                                                 Ws_self + (size_t)l * HID * HID,
                                                 Ws_neigh + (size_t)l * HID * HID,
                                                 biases + (size_t)l * HID,
                                                 gammas + (size_t)l * HID,
                                                 betas + (size_t)l * HID,
                                                 outp, nNodes);
    }
}